// GNNEncoder_81965155877636
// MI455X (gfx1250) — compile-verified
//
#include <hip/hip_runtime.h>

#define N_NODES 100000

typedef __attribute__((ext_vector_type(2))) float v2f;
typedef __attribute__((ext_vector_type(8))) float v8f;

// ---------------------------------------------------------------- zero scratch
__global__ void zero_f4(float4* __restrict__ p, long long n4) {
    long long i = (long long)blockIdx.x * blockDim.x + threadIdx.x;
    if (i < n4) p[i] = make_float4(0.f, 0.f, 0.f, 0.f);
}

// -------------------------------------------------- edge scatter (sum + degree)
// thread t -> edge e = t / D4, float4-group g = t % D4. Consecutive threads share
// an edge => coalesced float4 gather of x[src] and clustered f32 atomics on dst.
template<int D4, bool COUNT_DEG>
__global__ void scatter_add(const float* __restrict__ feat,
                            const int*  __restrict__ src,
                            const int*  __restrict__ dst,
                            float* __restrict__ agg,
                            float* __restrict__ deg,
                            long long total) {
    long long t = (long long)blockIdx.x * blockDim.x + threadIdx.x;
    if (t >= total) return;
    int e = (int)(t / D4);
    int g = (int)(t % D4);
    int s = src[e];
    int d = dst[e];
    const float4 v = ((const float4*)(feat + (long long)s * (D4 * 4)))[g];
    float* o = agg + (long long)d * (D4 * 4) + g * 4;
    atomicAdd(o + 0, v.x);
    atomicAdd(o + 1, v.y);
    atomicAdd(o + 2, v.z);
    atomicAdd(o + 3, v.w);
    if (COUNT_DEG && g == 0) atomicAdd(deg + d, 1.0f);
}

// ---------------------------------------------------------------- SAGE layer
// out[i,:] = act( (agg[i,:]/max(deg,1)) @ Wl^T + b + xin[i,:] @ Wr^T )
// Fused as [agg_scaled | xin] (16 x 2*DIN, staged in LDS) times [Wl | Wr]^T,
// computed with V_WMMA_F32_16X16X4_F32 chains. Block = 16 nodes, 8 waves each
// owning a 16-wide slice of the 128 output channels.
template<int DIN, bool RELU>
__global__ __launch_bounds__(256)
void sage_layer(const float* __restrict__ xin,   // [N, DIN]
                const float* __restrict__ agg,   // [N, DIN] (neighbor sums)
                const float* __restrict__ deg,   // [N]
                const float* __restrict__ Wl,    // [128, DIN]
                const float* __restrict__ Wr,    // [128, DIN]
                const float* __restrict__ bias,  // [128]
                float* __restrict__ out)         // [N, 128]
{
    constexpr int KTOT = 2 * DIN;
    constexpr int KPAD = KTOT + 2;               // +2 floats: bank-conflict pad
    __shared__ float Atile[16][KPAD];
    __shared__ float sscale[16];

    const int tid   = threadIdx.x;
    const int node0 = blockIdx.x * 16;

    if (tid < 16) sscale[tid] = 1.0f / fmaxf(deg[node0 + tid], 1.0f);
    __syncthreads();

    // Stage A = [agg * 1/deg | x] for 16 nodes.
    for (int idx = tid; idx < 16 * KTOT; idx += 256) {
        int row = idx / KTOT;
        int col = idx - row * KTOT;
        float v;
        if (col < DIN) v = agg[(long long)(node0 + row) * DIN + col] * sscale[row];
        else           v = xin[(long long)(node0 + row) * DIN + (col - DIN)];
        Atile[row][col] = v;
    }
    __syncthreads();

    const int lane = tid & 31;
    const int wave = tid >> 5;        // 0..7 -> output-channel tile
    const int l15  = lane & 15;
    const int half = lane >> 4;       // 0: K0/K1 (A) rows, 1: K2/K3
    const int ncol = wave * 16 + l15; // global output channel for this lane

    v8f acc;
    const float bv = bias[ncol];      // C[M][N] depends only on N
#pragma unroll
    for (int r = 0; r < 8; ++r) acc[r] = bv;

#pragma unroll 4
    for (int k = 0; k < KTOT; k += 4) {
        const int kk = k + half * 2;  // k<DIN iff kk<DIN (both 4-aligned chunks)
        v2f a;
        a.x = Atile[l15][kk];
        a.y = Atile[l15][kk + 1];
        const float* W = (kk < DIN) ? (Wl + (long long)ncol * DIN + kk)
                                    : (Wr + (long long)ncol * DIN + (kk - DIN));
        v2f b;
        b.x = W[0];
        b.y = W[1];
        acc = __builtin_amdgcn_wmma_f32_16x16x4_f32(false, a, false, b,
                                                    (short)0, acc, false, false);
    }

    // D layout: lanes 0-15 -> M = r, lanes 16-31 -> M = 8 + r; N = ncol.
#pragma unroll
    for (int r = 0; r < 8; ++r) {
        int row = node0 + r + half * 8;
        float v = acc[r];
        if (RELU) v = fmaxf(v, 0.0f);
        out[(long long)row * 128 + ncol] = v;
    }
}

// ---------------------------------------------------------------- launcher
extern "C" void kernel_launch(void* const* d_in, const int* in_sizes, int n_in,
                              void* d_out, int out_size, void* d_ws, size_t ws_size,
                              hipStream_t stream) {
    const float* x   = (const float*)d_in[0];
    const int*   ei  = (const int*)  d_in[1];   // int32 (JAX canonicalizes int64)
    const float* W1l = (const float*)d_in[2];
    const float* W1r = (const float*)d_in[3];
    const float* b1  = (const float*)d_in[4];
    const float* W2l = (const float*)d_in[5];
    const float* W2r = (const float*)d_in[6];
    const float* b2  = (const float*)d_in[7];
    float* out = (float*)d_out;

    const int E = in_sizes[1] / 2;              // 1,600,000
    const int* src = ei;
    const int* dst = ei + E;

    // Workspace layout (floats): deg [0,N) | agg1 [N, 65N) | agg2 [65N, 193N)
    float* ws   = (float*)d_ws;
    float* deg  = ws;
    float* agg1 = ws + (size_t)N_NODES;
    float* agg2 = ws + (size_t)N_NODES * 65;
    float* h    = out;  // layer-1 activations live in d_out; layer 2 overwrites
                        // each 16-row tile only after staging it into LDS.

    // 1) zero deg + agg1 + agg2 (193N floats, N divisible by 4)
    {
        long long n4 = (long long)N_NODES * 193 / 4;
        int blocks = (int)((n4 + 255) / 256);
        zero_f4<<<blocks, 256, 0, stream>>>((float4*)ws, n4);
    }

    // 2) scatter layer-1 messages + degrees: agg1[dst] += x[src], deg[dst] += 1
    {
        long long total = (long long)E * 16;    // 64 ch / 4
        int blocks = (int)((total + 255) / 256);
        scatter_add<16, true><<<blocks, 256, 0, stream>>>(x, src, dst, agg1, deg, total);
    }

    // 3) layer 1: h = relu([agg1/deg | x] @ [W1l|W1r]^T + b1)
    sage_layer<64, true><<<N_NODES / 16, 256, 0, stream>>>(x, agg1, deg,
                                                           W1l, W1r, b1, h);

    // 4) scatter layer-2 messages: agg2[dst] += h[src]
    {
        long long total = (long long)E * 32;    // 128 ch / 4
        int blocks = (int)((total + 255) / 256);
        scatter_add<32, false><<<blocks, 256, 0, stream>>>(h, src, dst, agg2, nullptr, total);
    }

    // 5) layer 2: out = [agg2/deg | h] @ [W2l|W2r]^T + b2
    sage_layer<128, false><<<N_NODES / 16, 256, 0, stream>>>(h, agg2, deg,
                                                             W2l, W2r, b2, out);
}